// DCRNN_266287972963
// MI455X (gfx1250) — compile-verified
//
#include <hip/hip_runtime.h>

// ---------------------------------------------------------------------------
// DCRNN cell forward, N=50000, E=800000, FIN=32, FOUT=64, K=3, H-state = 0.
// Math collapses to:
//   F = [X | T1o | T1i | T2o | T2i]  (N x 160)
//   [z|h] = F @ Wcat + [bz|bh]       (N x 128)  -> WMMA f32 16x16x4 GEMM
//   out   = (1 - sigmoid(z)) * tanh(h)
// R gate is dead (multiplies H=0). Only rows 0..31 of each (96,64) W matter.
// ---------------------------------------------------------------------------

typedef __attribute__((ext_vector_type(2))) float v2f;
typedef __attribute__((ext_vector_type(8))) float v8f;

#define N_NODES 50000
#define N_EDGES 800000
#define FIN 32
#define FOUT 64

// ---------------- small utility kernels ----------------

__global__ void fill_i32(int* p, int v, int n) {
    int i = blockIdx.x * blockDim.x + threadIdx.x;
    if (i < n) p[i] = v;
}

// foff[v] = lower_bound(src, v)  (src is sorted row-major by (src,dst))
__global__ void row_offsets_src(const int* __restrict__ src, int* __restrict__ foff,
                                int n, int E) {
    int v = blockIdx.x * blockDim.x + threadIdx.x;
    if (v > n) return;
    int lo = 0, hi = E;
    while (lo < hi) { int mid = (lo + hi) >> 1; if (src[mid] < v) lo = mid + 1; else hi = mid; }
    foff[v] = lo;
}

__global__ void count_dst(const int* __restrict__ dst, int* __restrict__ cnt, int E) {
    int e = blockIdx.x * blockDim.x + threadIdx.x;
    if (e < E) atomicAdd(&cnt[dst[e]], 1);
}

// single-workgroup exclusive scan (n = 50000, perf-irrelevant, deterministic)
__global__ void scan_excl(const int* __restrict__ cnt, int* __restrict__ offs, int n) {
    __shared__ int buf[1024];
    __shared__ int carry;
    if (threadIdx.x == 0) carry = 0;
    __syncthreads();
    for (int base = 0; base < n; base += 1024) {
        int i = base + (int)threadIdx.x;
        int v = (i < n) ? cnt[i] : 0;
        buf[threadIdx.x] = v;
        __syncthreads();
        for (int off = 1; off < 1024; off <<= 1) {
            int t = (threadIdx.x >= (unsigned)off) ? buf[threadIdx.x - off] : 0;
            __syncthreads();
            buf[threadIdx.x] += t;
            __syncthreads();
        }
        int incl = buf[threadIdx.x];
        int c = carry;
        __syncthreads();
        if (i < n) offs[i] = c + incl - v;
        if (threadIdx.x == 1023) carry = c + incl;
        __syncthreads();
    }
    if (threadIdx.x == 0) offs[n] = carry;
}

// scatter edges into transpose buckets (intra-bucket order fixed by sort below)
__global__ void place_edges(const int* __restrict__ dst, const int* __restrict__ toff,
                            int* __restrict__ cur, int* __restrict__ tedge, int E) {
    int e = blockIdx.x * blockDim.x + threadIdx.x;
    if (e >= E) return;
    int d = dst[e];
    int pos = toff[d] + atomicAdd(&cur[d], 1);
    tedge[pos] = e;
}

// per-bucket insertion sort by original edge id (== sort by (dst,src)): deterministic
__global__ void sort_buckets(int* __restrict__ tedge, const int* __restrict__ toff, int n) {
    int d = blockIdx.x * blockDim.x + threadIdx.x;
    if (d >= n) return;
    int s = toff[d], e = toff[d + 1];
    for (int i = s + 1; i < e; ++i) {
        int key = tedge[i];
        int j = i - 1;
        while (j >= s && tedge[j] > key) { tedge[j + 1] = tedge[j]; --j; }
        tedge[j + 1] = key;
    }
}

__global__ void inv_perm(const int* __restrict__ tedge, int* __restrict__ q, int E) {
    int j = blockIdx.x * blockDim.x + threadIdx.x;
    if (j < E) q[tedge[j]] = j;
}

// deterministic sequential degree sums per node
__global__ void degrees(const int* __restrict__ foff, const int* __restrict__ toff,
                        const int* __restrict__ tedge, const float* __restrict__ ew,
                        float* __restrict__ dout, float* __restrict__ din, int n) {
    int v = blockIdx.x * blockDim.x + threadIdx.x;
    if (v >= n) return;
    float so = 0.f;
    for (int f = foff[v]; f < foff[v + 1]; ++f) so += ew[f];
    dout[v] = so;
    float si = 0.f;
    for (int j = toff[v]; j < toff[v + 1]; ++j) si += ew[tedge[j]];
    din[v] = si;
}

// wfor[e] = 1/deg_out[src[e]]   (forward-direction norm, used in transpose gather)
// wrev[f] = 1/deg_in[src[q[f]]] (faithful to reference's norm_in[e] with e=q[f])
__global__ void edge_norms(const int* __restrict__ src, const int* __restrict__ q,
                           const float* __restrict__ dout, const float* __restrict__ din,
                           float* __restrict__ wfor, float* __restrict__ wrev, int E) {
    int e = blockIdx.x * blockDim.x + threadIdx.x;
    if (e >= E) return;
    wfor[e] = 1.0f / dout[src[e]];
    wrev[e] = 1.0f / din[src[q[e]]];
}

// ---------------- SpMM: one wave per node, lane = channel (wave32) ----------------
// out[v,:] = alpha * sum_{j in [offs[v],offs[v+1])} wgt[f] * in[nbr[f],:]  - (Xsub? X[v,:] : 0)
// with f = eid ? eid[j] : j.  Gather-based -> deterministic, coalesced 128B row reads.
__global__ void spmm_gather(float* __restrict__ out, const float* __restrict__ in,
                            const float* __restrict__ Xsub,
                            const int* __restrict__ offs, const int* __restrict__ eid,
                            const int* __restrict__ nbr, const float* __restrict__ wgt,
                            float alpha, int n) {
    int gt = blockIdx.x * blockDim.x + threadIdx.x;
    int v = gt >> 5;
    int c = threadIdx.x & 31;
    if (v >= n) return;
    float acc = 0.f;
    int s = offs[v], e = offs[v + 1];
    for (int j = s; j < e; ++j) {
        int f = eid ? eid[j] : j;
        float w = wgt[f];
        int nb = nbr[f];
        acc += w * in[(size_t)nb * FIN + c];
    }
    float r = alpha * acc;
    if (Xsub) r -= Xsub[(size_t)v * FIN + c];
    out[(size_t)v * FIN + c] = r;
}

// ---------------- weight repack: Wcat[160][128], cols 0..63 = z gate, 64..127 = h ----------
// row blocks: 0:X (W[0,0]+W[1,0]), 1:T1o (W[0,1]), 2:T1i (W[1,1]), 3:T2o (W[0,2]), 4:T2i (W[1,2])
// only cin rows 0..31 of each (96,64) slice are live (H-state = 0).
__global__ void repack_w(const float* __restrict__ Wz, const float* __restrict__ Wh,
                         float* __restrict__ Wcat) {
    int idx = blockIdx.x * blockDim.x + threadIdx.x;
    if (idx >= 160 * 128) return;
    int row = idx >> 7, col = idx & 127;
    int b = row >> 5, c = row & 31;
    const float* W = (col < 64) ? Wz : Wh;
    int o = col & 63;
    // W layout: [2][3][96][64]
    #define WAT(i, k) W[((((size_t)(i)) * 3 + (k)) * 96 + c) * 64 + o]
    float v;
    switch (b) {
        case 0:  v = WAT(0, 0) + WAT(1, 0); break;
        case 1:  v = WAT(0, 1); break;
        case 2:  v = WAT(1, 1); break;
        case 3:  v = WAT(0, 2); break;
        default: v = WAT(1, 2); break;
    }
    #undef WAT
    Wcat[idx] = v;
}

// ---------------- fused gate GEMM via V_WMMA_F32_16X16X4_F32 ----------------
// One wave computes a 16-row strip x 128 cols (8 accumulator tiles).
// A frag (16x4 f32): lanes 0-15 -> K{kb,kb+1}, lanes 16-31 -> K{kb+2,kb+3} (ISA 7.12.2)
// B frag (4x16 f32): row-striped across lanes within a VGPR (same half-split).
// C/D (16x16 f32): VGPR r: lanes 0-15 = row r, lanes 16-31 = row r+8.
__global__ __launch_bounds__(128) void gemm_gates(
        const float* __restrict__ X, const float* __restrict__ T1o,
        const float* __restrict__ T1i, const float* __restrict__ T2o,
        const float* __restrict__ T2i, const float* __restrict__ Wcat,
        const float* __restrict__ bz, const float* __restrict__ bh,
        float* __restrict__ out, int nstrips) {
    int wave = threadIdx.x >> 5;
    int lane = threadIdx.x & 31;
    int strip = blockIdx.x * 4 + wave;          // wave-uniform guard: EXEC stays all-ones
    if (strip >= nstrips) return;
    int m0 = strip * 16;
    int hl = lane >> 4;                          // half select (K pair / row+8)
    int l15 = lane & 15;

    const float* blk[5] = {X, T1o, T1i, T2o, T2i};
    v8f acc[8];
    #pragma unroll
    for (int t = 0; t < 8; ++t) acc[t] = (v8f)0.0f;

    #pragma unroll
    for (int b = 0; b < 5; ++b) {
        const float* A = blk[b];
        #pragma unroll
        for (int kk = 0; kk < FIN; kk += 4) {
            int kb = b * FIN + kk;
            const float* ap = A + (size_t)(m0 + l15) * FIN + kk + 2 * hl;
            v2f a;
            a.x = ap[0];
            a.y = ap[1];
            const float* brow0 = Wcat + (size_t)(kb + 2 * hl) * 128 + l15;
            #pragma unroll
            for (int t = 0; t < 8; ++t) {
                v2f bfrag;
                bfrag.x = brow0[t * 16];
                bfrag.y = brow0[t * 16 + 128];
                acc[t] = __builtin_amdgcn_wmma_f32_16x16x4_f32(
                    false, a, false, bfrag, (short)0, acc[t], false, false);
            }
        }
    }

    // epilogue: z = tiles 0..3, h = tiles 4..7 ; out = (1-sigmoid(z))*tanh(h)
    #pragma unroll
    for (int t = 0; t < 4; ++t) {
        int col = t * 16 + l15;
        float bzv = bz[col];
        float bhv = bh[col];
        #pragma unroll
        for (int r = 0; r < 8; ++r) {
            int row = m0 + r + 8 * hl;
            float z = acc[t][r] + bzv;
            float h = acc[t + 4][r] + bhv;
            float sz = 1.0f / (1.0f + __expf(-z));
            float th = tanhf(h);
            out[(size_t)row * FOUT + col] = (1.0f - sz) * th;
        }
    }
}

// ---------------------------------------------------------------------------

static inline size_t align256(size_t x) { return (x + 255) & ~(size_t)255; }

extern "C" void kernel_launch(void* const* d_in, const int* in_sizes, int n_in,
                              void* d_out, int out_size, void* d_ws, size_t ws_size,
                              hipStream_t stream) {
    const int n = N_NODES, E = N_EDGES;
    const float* X   = (const float*)d_in[0];
    const int*   src = (const int*)d_in[1];
    const int*   dst = src + E;
    const float* ew  = (const float*)d_in[2];
    const float* Wz  = (const float*)d_in[3];
    const float* bz  = (const float*)d_in[4];
    // d_in[5]=Wr, d_in[6]=br : dead (R gate multiplies H=0)
    const float* Wh  = (const float*)d_in[7];
    const float* bh  = (const float*)d_in[8];
    float* out = (float*)d_out;

    // workspace carve
    char* ws = (char*)d_ws;
    size_t off = 0;
    #define CARVE(ty, name, count) ty* name = (ty*)(ws + off); off = align256(off + (size_t)(count) * sizeof(ty))
    CARVE(float, T1o, (size_t)n * FIN);
    CARVE(float, T1i, (size_t)n * FIN);
    CARVE(float, T2o, (size_t)n * FIN);
    CARVE(float, T2i, (size_t)n * FIN);
    CARVE(int,   foff, n + 1);
    CARVE(int,   toff, n + 1);
    CARVE(int,   cnt,  n);
    CARVE(int,   cur,  n);
    CARVE(int,   tedge, E);
    CARVE(int,   qinv,  E);
    CARVE(float, wfor,  E);
    CARVE(float, wrev,  E);
    CARVE(float, dout,  n);
    CARVE(float, din,   n);
    CARVE(float, Wcat,  160 * 128);
    #undef CARVE
    (void)ws_size; (void)in_sizes; (void)n_in; (void)out_size;

    const int TB = 256;
    dim3 blk(TB);

    // 1) transpose structure (deterministic after per-bucket sort)
    fill_i32<<<dim3((n + TB - 1) / TB), blk, 0, stream>>>(cnt, 0, n);
    fill_i32<<<dim3((n + TB - 1) / TB), blk, 0, stream>>>(cur, 0, n);
    count_dst<<<dim3((E + TB - 1) / TB), blk, 0, stream>>>(dst, cnt, E);
    scan_excl<<<dim3(1), dim3(1024), 0, stream>>>(cnt, toff, n);
    row_offsets_src<<<dim3((n + 1 + TB - 1) / TB), blk, 0, stream>>>(src, foff, n, E);
    place_edges<<<dim3((E + TB - 1) / TB), blk, 0, stream>>>(dst, toff, cur, tedge, E);
    sort_buckets<<<dim3((n + TB - 1) / TB), blk, 0, stream>>>(tedge, toff, n);
    inv_perm<<<dim3((E + TB - 1) / TB), blk, 0, stream>>>(tedge, qinv, E);

    // 2) degrees + per-edge norms (faithful: both norms indexed by src)
    degrees<<<dim3((n + TB - 1) / TB), blk, 0, stream>>>(foff, toff, tedge, ew, dout, din, n);
    edge_norms<<<dim3((E + TB - 1) / TB), blk, 0, stream>>>(src, qinv, dout, din, wfor, wrev, E);

    // 3) diffusion features (gather SpMM, wave-per-node)
    int spmmBlocks = (n * 32 + TB - 1) / TB;
    // T1o[d] = sum_{e:dst[e]=d} wfor[e] * X[src[e]]
    spmm_gather<<<dim3(spmmBlocks), blk, 0, stream>>>(T1o, X, nullptr, toff, tedge, src, wfor, 1.0f, n);
    // T1i[v] = sum_{f:src[f]=v} wrev[f] * X[dst[f]]
    spmm_gather<<<dim3(spmmBlocks), blk, 0, stream>>>(T1i, X, nullptr, foff, nullptr, dst, wrev, 1.0f, n);
    // T2o = 2*A_out*T1o - X ; T2i = 2*A_in*T1i - X
    spmm_gather<<<dim3(spmmBlocks), blk, 0, stream>>>(T2o, T1o, X, toff, tedge, src, wfor, 2.0f, n);
    spmm_gather<<<dim3(spmmBlocks), blk, 0, stream>>>(T2i, T1i, X, foff, nullptr, dst, wrev, 2.0f, n);

    // 4) fused dual-gate GEMM + activation epilogue (WMMA f32 16x16x4)
    repack_w<<<dim3((160 * 128 + TB - 1) / TB), blk, 0, stream>>>(Wz, Wh, Wcat);
    int nstrips = n / 16;                        // 3125 (N divisible by 16)
    int gemmBlocks = (nstrips + 3) / 4;          // 4 waves per block
    gemm_gates<<<dim3(gemmBlocks), dim3(128), 0, stream>>>(
        X, T1o, T1i, T2o, T2i, Wcat, bz, bh, out, nstrips);
}